// NodeFeatures_37967510896745
// MI455X (gfx1250) — compile-verified
//
#include <hip/hip_runtime.h>

// ---------------------------------------------------------------------------
// Problem: B=8, N=256, H=128
//   Ux = 0.5*(x @ U_w^T + U_b)          (2048x128 @ 128x128 GEMM)  -> WMMA
//   Vx = 0.5*(x @ V_w^T + V_b)          (2048x128 @ 128x128 GEMM)  -> WMMA
//   agg[b,i,h]   = sum_j edge_gate[b,i,j,h] * Vx[b,j,h]            -> stream
//   denom[b,i,h] = 1e-20 + sum_j edge_gate[b,i,j,h]                -> stream
//   out = Ux + agg/denom
// edge_gate = 256 MB dominates: ~11 us at 23.3 TB/s. Kernel 2 is a single
// fully-coalesced pass over it. Kernel 1 uses v_wmma_f32_16x16x4_f32.
// ---------------------------------------------------------------------------

typedef float v2f __attribute__((ext_vector_type(2)));
typedef float v8f __attribute__((ext_vector_type(8)));

#define BB 8
#define NN 256
#define HH 128

// One wave per 16x16 output tile. K-loop: 32 x v_wmma_f32_16x16x4_f32.
// Computes out = 0.5*(x @ W^T + bias) for both U (->d_out) and V (->d_ws).
__global__ __launch_bounds__(32) void nf_gemm_wmma_f32(
    const float* __restrict__ x,
    const float* __restrict__ Uw, const float* __restrict__ Ub,
    const float* __restrict__ Vw, const float* __restrict__ Vb,
    float* __restrict__ outU, float* __restrict__ outV)
{
    const int lane  = threadIdx.x;        // 0..31
    const int mTile = blockIdx.x;         // 0..127 (M = B*N = 2048)
    const int nTile = blockIdx.y;         // 0..7   (N = H = 128)
    const int which = blockIdx.z;         // 0 = U, 1 = V

    const float* __restrict__ W    = which ? Vw : Uw;
    const float* __restrict__ bias = which ? Vb : Ub;
    float*       __restrict__ outp = which ? outV : outU;

    // ISA 32-bit A 16x4 layout: lanes 0-15 hold rows M=0..15 with K={0,1},
    // lanes 16-31 hold the same rows with K={2,3}. B 4x16 is symmetric.
    const int m    = mTile * 16 + (lane & 15);
    const int n    = nTile * 16 + (lane & 15);
    const int kSel = (lane >> 4) << 1;    // 0 or 2

    const float* __restrict__ aRow = x + m * HH;   // A row (x is row-major MxK)
    const float* __restrict__ bRow = W + n * HH;   // B col n = W row n (W^T)

    v8f c = {0.f, 0.f, 0.f, 0.f, 0.f, 0.f, 0.f, 0.f};

#pragma unroll
    for (int k = 0; k < HH; k += 4) {
        v2f a, b;
        a.x = aRow[k + kSel];
        a.y = aRow[k + kSel + 1];
        b.x = bRow[k + kSel];
        b.y = bRow[k + kSel + 1];
        // 8 args: (neg_a, A, neg_b, B, c_mod, C, reuse_a, reuse_b)
        c = __builtin_amdgcn_wmma_f32_16x16x4_f32(
                false, a, false, b, (short)0, c, false, false);
    }

    // C/D layout: VGPR r holds M = r (lanes 0-15) / M = r+8 (lanes 16-31).
    const int   col     = nTile * 16 + (lane & 15);
    const float bv      = bias[col];
    const int   rowBase = mTile * 16 + ((lane >> 4) << 3);
#pragma unroll
    for (int r = 0; r < 8; ++r) {
        outp[(size_t)(rowBase + r) * HH + col] = (c[r] + bv) * 0.5f;
    }
}

// One 256-thread block per (b,i). 8 j-groups x 32 lanes; each group streams
// one contiguous 512B edge_gate row per iteration as float4 (b128 loads).
__global__ __launch_bounds__(256) void nf_agg_stream(
    const float* __restrict__ edge_gate,
    const float* __restrict__ Vx,
    float*       __restrict__ out)   // holds Ux on entry; final result on exit
{
    const int bi   = blockIdx.x;          // 0..2047
    const int b    = bi >> 8;
    const int i    = bi & (NN - 1);
    const int tid  = threadIdx.x;
    const int g    = tid >> 5;            // j-group 0..7
    const int lane = tid & 31;            // h-chunk: h = lane*4

    const float4* __restrict__ egBase =
        (const float4*)(edge_gate + ((size_t)b * NN + i) * (size_t)NN * HH);
    const float4* __restrict__ vxBase =
        (const float4*)(Vx + (size_t)b * NN * HH);

    float4 acc = make_float4(0.f, 0.f, 0.f, 0.f);
    float4 den = make_float4(0.f, 0.f, 0.f, 0.f);

    for (int j = g; j < NN; j += 8) {
        if (j + 16 < NN) {
            __builtin_prefetch(&egBase[(size_t)(j + 16) * 32 + lane], 0, 0);
        }
        const float4 eg = egBase[(size_t)j * 32 + lane];
        const float4 vx = vxBase[(size_t)j * 32 + lane];   // L2-hot (128 KB/batch)
        acc.x = fmaf(eg.x, vx.x, acc.x);
        acc.y = fmaf(eg.y, vx.y, acc.y);
        acc.z = fmaf(eg.z, vx.z, acc.z);
        acc.w = fmaf(eg.w, vx.w, acc.w);
        den.x += eg.x; den.y += eg.y; den.z += eg.z; den.w += eg.w;
    }

    __shared__ float4 sAcc[8][32];
    __shared__ float4 sDen[8][32];
    sAcc[g][lane] = acc;
    sDen[g][lane] = den;
    __syncthreads();

    if (g == 0) {
        float4 A = sAcc[0][lane];
        float4 D = sDen[0][lane];
#pragma unroll
        for (int gg = 1; gg < 8; ++gg) {
            const float4 a2 = sAcc[gg][lane];
            const float4 d2 = sDen[gg][lane];
            A.x += a2.x; A.y += a2.y; A.z += a2.z; A.w += a2.w;
            D.x += d2.x; D.y += d2.y; D.z += d2.z; D.w += d2.w;
        }
        const size_t o = ((size_t)b * NN + i) * HH + (size_t)lane * 4;
        out[o + 0] += A.x / (1e-20f + D.x);
        out[o + 1] += A.y / (1e-20f + D.y);
        out[o + 2] += A.z / (1e-20f + D.z);
        out[o + 3] += A.w / (1e-20f + D.w);
    }
}

extern "C" void kernel_launch(void* const* d_in, const int* in_sizes, int n_in,
                              void* d_out, int out_size, void* d_ws, size_t ws_size,
                              hipStream_t stream) {
    // setup_inputs order: x, edge_gate, adjacency, PN, NN, U_w, U_b, V_w, V_b
    const float* x  = (const float*)d_in[0];
    const float* eg = (const float*)d_in[1];
    const float* Uw = (const float*)d_in[5];
    const float* Ub = (const float*)d_in[6];
    const float* Vw = (const float*)d_in[7];
    const float* Vb = (const float*)d_in[8];
    float* out = (float*)d_out;
    float* Vx  = (float*)d_ws;          // B*N*H floats = 1 MB scratch

    // Kernel 1: Ux -> d_out, Vx -> d_ws (WMMA f32 GEMMs)
    dim3 g1(128, 8, 2);
    nf_gemm_wmma_f32<<<g1, 32, 0, stream>>>(x, Uw, Ub, Vw, Vb, out, Vx);

    // Kernel 2: stream 256 MB of edge_gate once; out = Ux + agg/denom
    nf_agg_stream<<<BB * NN, 256, 0, stream>>>(eg, Vx, out);
}